// Tempo_80831284511211
// MI455X (gfx1250) — compile-verified
//
#include <hip/hip_runtime.h>
#include <math.h>

// Model dims (compile-time, match reference)
static constexpr int kB = 2;
static constexpr int kS = 2048;
static constexpr int kD = 512;
static constexpr int kL = 4;
static constexpr int kV = 32000;
static constexpr int kBS = kB * kS;              // 4096 rows
static constexpr size_t kNBD = (size_t)kBS * kD; // 2,097,152 elems

typedef __attribute__((ext_vector_type(2))) float v2f;
typedef __attribute__((ext_vector_type(8))) float v8f;

__device__ __forceinline__ float gelu_exact(float x) {
    return 0.5f * x * (1.0f + erff(x * 0.70710678118654752440f));
}
__device__ __forceinline__ float sigmoidf(float x) {
    return 1.0f / (1.0f + expf(-x));
}

// ---------------------------------------------------------------------------
// Embedding: x[b,s,:] = emb[tok[b,s],:] + pos[s,:]
// ---------------------------------------------------------------------------
__global__ void embed_kernel(const float* __restrict__ emb,
                             const float* __restrict__ pos,
                             const int*   __restrict__ tokens,
                             float* __restrict__ x) {
    size_t idx = (size_t)blockIdx.x * blockDim.x + threadIdx.x; // < B*S*D
    int d = (int)(idx % kD);
    int bs = (int)(idx / kD);
    int s = bs % kS;
    int tok = tokens[bs];
    x[idx] = emb[(size_t)tok * kD + d] + pos[(size_t)s * kD + d];
}

// ---------------------------------------------------------------------------
// LayerNorm: one block per row, runtime width W (512 or 2048)
// ---------------------------------------------------------------------------
__global__ void layernorm_kernel(const float* __restrict__ x,
                                 const float* __restrict__ g,
                                 const float* __restrict__ b,
                                 float* __restrict__ y, int W) {
    const int row = blockIdx.x;
    const int tid = threadIdx.x;
    const float* xr = x + (size_t)row * W;
    float* yr = y + (size_t)row * W;
    __shared__ float red[256];

    float s = 0.f;
    for (int i = tid; i < W; i += 256) s += xr[i];
    red[tid] = s; __syncthreads();
    for (int o = 128; o > 0; o >>= 1) { if (tid < o) red[tid] += red[tid + o]; __syncthreads(); }
    const float mean = red[0] / (float)W;
    __syncthreads();

    float v = 0.f;
    for (int i = tid; i < W; i += 256) { float d2 = xr[i] - mean; v += d2 * d2; }
    red[tid] = v; __syncthreads();
    for (int o = 128; o > 0; o >>= 1) { if (tid < o) red[tid] += red[tid + o]; __syncthreads(); }
    const float rs = rsqrtf(red[0] / (float)W + 1e-5f);

    for (int i = tid; i < W; i += 256)
        yr[i] = (xr[i] - mean) * rs * g[i] + b[i];
}

// ---------------------------------------------------------------------------
// FP32 WMMA GEMM: C[M,N] = act(A[M,K] @ B + bias) (+ res1 + res2)
//   TRANSB == 0: B is [K,N] row-major.  TRANSB == 1: B is [N,K] row-major.
// Block: 256 thr (8 waves). Macro tile 64(M) x 128(N); wave tile 32x32
// (four 16x16 f32 accumulators -> 4 WMMA per 4 fragment loads).
// K staged through double-buffered LDS in slices of 16 (4 WMMA K-steps),
// with register prefetch of the next stage: one barrier per stage.
// Requires M%64==0, N%128==0, K%16==0 (true for all calls here).
// ---------------------------------------------------------------------------
#define ACT_NONE 0
#define ACT_GELU 1

static constexpr int kLdsA = 20;   // A row stride (floats): mult of 4, 20m%64 distinct
static constexpr int kLdsB = 132;  // B row stride (floats): mult of 4

template <int TRANSB>
__global__ __launch_bounds__(256)
void gemm_f32_wmma(const float* __restrict__ A,
                   const float* __restrict__ Bm,
                   const float* __restrict__ bias,
                   const float* __restrict__ res1,
                   const float* __restrict__ res2,
                   float* __restrict__ C,
                   int M, int N, int K, int act) {
    const int bm = blockIdx.y * 64;
    const int bn = blockIdx.x * 128;
    const int tid  = threadIdx.x;
    const int lane = tid & 31;
    const int w    = tid >> 5;
    const int wm = (w & 1) * 32;   // wave row offset in macro tile
    const int wn = (w >> 1) * 32;  // wave col offset in macro tile

    __shared__ float sA[2][64 * kLdsA];  // 64 rows x 16 k
    __shared__ float sB[2][16 * kLdsB];  // 16 k x 128 cols

    v8f acc00 = {}, acc01 = {}, acc10 = {}, acc11 = {};
    const int mrow = lane & 15;    // M index for A frag / N index for B frag
    const int kh   = lane >> 4;    // lane-half selects K pair (A) / row pair (B)

    // staging thread mapping
    const int ar = tid >> 2, ac = (tid & 3) * 4;    // A: 64 rows x 4 float4 segs
    const int br = tid >> 4, bc = (tid & 15) * 4;   // B: 16 rows x 16 segs (x2 halves)
    const int tc = tid >> 1, tr = (tid & 1) * 8;    // B^T: 128 cols x 2 k-octets

    float4 pa, pb0, pb1;
    auto load_stage = [&](int k0) {
        pa = *(const float4*)(A + (size_t)(bm + ar) * K + k0 + ac);
        if (!TRANSB) {
            pb0 = *(const float4*)(Bm + (size_t)(k0 + br) * N + bn + bc);
            pb1 = *(const float4*)(Bm + (size_t)(k0 + br) * N + bn + bc + 64);
        } else {
            pb0 = *(const float4*)(Bm + (size_t)(bn + tc) * K + k0 + tr);
            pb1 = *(const float4*)(Bm + (size_t)(bn + tc) * K + k0 + tr + 4);
        }
    };
    auto store_stage = [&](int buf) {
        *(float4*)&sA[buf][ar * kLdsA + ac] = pa;
        if (!TRANSB) {
            *(float4*)&sB[buf][br * kLdsB + bc] = pb0;
            *(float4*)&sB[buf][br * kLdsB + bc + 64] = pb1;
        } else {
            sB[buf][(tr + 0) * kLdsB + tc] = pb0.x;
            sB[buf][(tr + 1) * kLdsB + tc] = pb0.y;
            sB[buf][(tr + 2) * kLdsB + tc] = pb0.z;
            sB[buf][(tr + 3) * kLdsB + tc] = pb0.w;
            sB[buf][(tr + 4) * kLdsB + tc] = pb1.x;
            sB[buf][(tr + 5) * kLdsB + tc] = pb1.y;
            sB[buf][(tr + 6) * kLdsB + tc] = pb1.z;
            sB[buf][(tr + 7) * kLdsB + tc] = pb1.w;
        }
    };

    load_stage(0);
    store_stage(0);
    __syncthreads();

    int buf = 0;
    for (int k0 = 0; k0 < K; k0 += 16) {
        const bool more = (k0 + 16) < K;
        if (more) load_stage(k0 + 16);   // prefetch next stage into registers

        const float* __restrict__ pA = &sA[buf][0];
        const float* __restrict__ pB = &sB[buf][0];
        #pragma unroll
        for (int kk = 0; kk < 4; ++kk) {
            const int ka = kk * 4 + 2 * kh;
            v2f a0, a1, b0, b1;
            a0.x = pA[(wm + mrow) * kLdsA + ka];
            a0.y = pA[(wm + mrow) * kLdsA + ka + 1];
            a1.x = pA[(wm + 16 + mrow) * kLdsA + ka];
            a1.y = pA[(wm + 16 + mrow) * kLdsA + ka + 1];
            b0.x = pB[ka * kLdsB + wn + mrow];
            b0.y = pB[(ka + 1) * kLdsB + wn + mrow];
            b1.x = pB[ka * kLdsB + wn + 16 + mrow];
            b1.y = pB[(ka + 1) * kLdsB + wn + 16 + mrow];
            acc00 = __builtin_amdgcn_wmma_f32_16x16x4_f32(
                false, a0, false, b0, (short)0, acc00, false, false);
            acc01 = __builtin_amdgcn_wmma_f32_16x16x4_f32(
                false, a0, false, b1, (short)0, acc01, false, false);
            acc10 = __builtin_amdgcn_wmma_f32_16x16x4_f32(
                false, a1, false, b0, (short)0, acc10, false, false);
            acc11 = __builtin_amdgcn_wmma_f32_16x16x4_f32(
                false, a1, false, b1, (short)0, acc11, false, false);
        }
        if (more) store_stage(buf ^ 1);  // fill the other buffer
        __syncthreads();
        buf ^= 1;
    }

    // Epilogue: C/D layout -> VGPR j: lanes0-15 M=j, lanes16-31 M=j+8
    auto store_acc = [&](const v8f& acc, int mt, int nt) {
        const int col = bn + wn + nt * 16 + mrow;
        #pragma unroll
        for (int j = 0; j < 8; ++j) {
            const int row = bm + wm + mt * 16 + j + kh * 8;
            float v = acc[j];
            if (bias) v += bias[col];
            if (act == ACT_GELU) v = gelu_exact(v);
            const size_t o = (size_t)row * N + col;
            if (res1) v += res1[o];
            if (res2) v += res2[o];
            C[o] = v;
        }
    };
    store_acc(acc00, 0, 0);
    store_acc(acc01, 0, 1);
    store_acc(acc10, 1, 0);
    store_acc(acc11, 1, 1);
}

// ---------------------------------------------------------------------------
// TPI scan: one thread per (b,d) lane, serial over s (true data dependence),
// d-contiguous across threads => coalesced. Writes ctx [B,S,4D] directly.
// ---------------------------------------------------------------------------
__global__ void tpi_scan_kernel(const float* __restrict__ h,
                                const float* __restrict__ omega,
                                const float* __restrict__ m_pre,
                                const float* __restrict__ g_pre,
                                const float* __restrict__ q,
                                const float* __restrict__ phi_init,
                                const float* __restrict__ int_scale,
                                float* __restrict__ ctx) {
    const int t = blockIdx.x * blockDim.x + threadIdx.x;  // < B*D
    const int b = t / kD;
    const int d = t % kD;
    const float isc = fabsf(int_scale[d]);

    float phi_acc = 0.f, mr = 0.f, mi = 0.f, mag_acc = 0.f;
    for (int s = 0; s < kS; ++s) {
        const size_t off = ((size_t)b * kS + s) * kD + d;
        const float hv   = h[off];
        const float gate = sigmoidf(g_pre[off]);
        const float mag  = sigmoidf(m_pre[off]) * 5.0f;
        phi_acc += gate * omega[off] * isc;
        const float phi = phi_init[off] + phi_acc;
        float cp, sp; sincosf(phi, &sp, &cp);
        const float wc = mag * hv;
        mr += wc * cp; mi += wc * sp; mag_acc += mag;
        const float inv = rsqrtf(mag_acc + 1e-8f);
        const float mrn = mr * inv, min_ = mi * inv;
        const float pq = phi + q[off];
        float cq, sq; sincosf(pq, &sq, &cq);
        const size_t o4 = ((size_t)b * kS + s) * (4 * kD) + d;
        ctx[o4]            = hv * cp;
        ctx[o4 + kD]       = hv * sp;
        ctx[o4 + 2 * kD]   = mrn * cq + min_ * sq;
        ctx[o4 + 3 * kD]   = min_ * cq - mrn * sq;
    }
}

// ---------------------------------------------------------------------------
static inline void launch_gemm(const float* A, const float* Bm, const float* bias,
                               const float* res1, const float* res2, float* C,
                               int M, int N, int K, int transB, int act,
                               hipStream_t stream) {
    dim3 grid(N / 128, M / 64);
    if (transB)
        gemm_f32_wmma<1><<<grid, 256, 0, stream>>>(A, Bm, bias, res1, res2, C, M, N, K, act);
    else
        gemm_f32_wmma<0><<<grid, 256, 0, stream>>>(A, Bm, bias, res1, res2, C, M, N, K, act);
}

extern "C" void kernel_launch(void* const* d_in, const int* in_sizes, int n_in,
                              void* d_out, int out_size, void* d_ws, size_t ws_size,
                              hipStream_t stream) {
    const float* emb    = (const float*)d_in[0];
    const float* pos    = (const float*)d_in[1];
    const float* ln1_g  = (const float*)d_in[2];
    const float* ln1_b  = (const float*)d_in[3];
    const float* W_om   = (const float*)d_in[4];
    const float* b_om   = (const float*)d_in[5];
    const float* W_p1   = (const float*)d_in[6];
    const float* b_p1   = (const float*)d_in[7];
    const float* W_p2   = (const float*)d_in[8];
    const float* b_p2   = (const float*)d_in[9];
    const float* W_g    = (const float*)d_in[10];
    const float* b_g    = (const float*)d_in[11];
    const float* i_scl  = (const float*)d_in[12];
    const float* W_m    = (const float*)d_in[13];
    const float* b_m    = (const float*)d_in[14];
    const float* W_q    = (const float*)d_in[15];
    const float* b_q    = (const float*)d_in[16];
    const float* ln2_g  = (const float*)d_in[17];
    const float* ln2_b  = (const float*)d_in[18];
    const float* W_o1   = (const float*)d_in[19];
    const float* b_o1   = (const float*)d_in[20];
    const float* W_o2   = (const float*)d_in[21];
    const float* b_o2   = (const float*)d_in[22];
    const float* lnf_g  = (const float*)d_in[23];
    const float* lnf_b  = (const float*)d_in[24];
    const int*   tokens = (const int*)d_in[25];
    float* out = (float*)d_out;
    float* ws  = (float*)d_ws;

    // Workspace layout (units of kNBD = B*S*D floats); peak 16*kNBD = 128 MB
    float* x     = ws;                 // [BS, D]
    float* h     = ws + 1 * kNBD;      // [BS, D]
    float* g_om  = ws + 2 * kNBD;      // omega
    float* g_m   = ws + 3 * kNBD;      // magnitude pre-act
    float* g_g   = ws + 4 * kNBD;      // gate pre-act
    float* g_q   = ws + 5 * kNBD;      // query phase
    float* g_p1  = ws + 6 * kNBD;      // gelu(h@W_p1+b)
    float* g_phi = ws + 7 * kNBD;      // phi_init
    float* ctx   = ws + 8 * kNBD;      // [BS, 4D]
    float* cnorm = ws + 12 * kNBD;     // [BS, 4D]
    float* o1    = ws + 2 * kNBD;      // [BS, 2D] (reuses dead g_om/g_m)

    // 1) embedding + positional
    embed_kernel<<<(int)(kNBD / 256), 256, 0, stream>>>(emb, pos, tokens, x);

    for (int i = 0; i < kL; ++i) {
        const size_t wdd = (size_t)i * kD * kD;
        // h = LN1(x)
        layernorm_kernel<<<kBS, 256, 0, stream>>>(x, ln1_g + i * kD, ln1_b + i * kD, h, kD);
        // six D x D projections of h
        launch_gemm(h, W_om + wdd, b_om + i * kD, nullptr, nullptr, g_om,
                    kBS, kD, kD, 0, ACT_NONE, stream);
        launch_gemm(h, W_m + wdd, b_m + i * kD, nullptr, nullptr, g_m,
                    kBS, kD, kD, 0, ACT_NONE, stream);
        launch_gemm(h, W_g + wdd, b_g + i * kD, nullptr, nullptr, g_g,
                    kBS, kD, kD, 0, ACT_NONE, stream);
        launch_gemm(h, W_q + wdd, b_q + i * kD, nullptr, nullptr, g_q,
                    kBS, kD, kD, 0, ACT_NONE, stream);
        launch_gemm(h, W_p1 + wdd, b_p1 + i * kD, nullptr, nullptr, g_p1,
                    kBS, kD, kD, 0, ACT_GELU, stream);
        launch_gemm(g_p1, W_p2 + wdd, b_p2 + i * kD, nullptr, nullptr, g_phi,
                    kBS, kD, kD, 0, ACT_NONE, stream);
        // phase-interference scan -> ctx [BS, 4D]
        tpi_scan_kernel<<<(kB * kD) / 64, 64, 0, stream>>>(
            h, g_om, g_m, g_g, g_q, g_phi, i_scl + i * kD, ctx);
        // LN2 over 4D
        layernorm_kernel<<<kBS, 256, 0, stream>>>(ctx, ln2_g + (size_t)i * 4 * kD,
                                                  ln2_b + (size_t)i * 4 * kD, cnorm, 4 * kD);
        // MLP: 4D -> 2D (gelu) -> D, fused residual x += h + out
        launch_gemm(cnorm, W_o1 + (size_t)i * 4 * kD * 2 * kD, b_o1 + (size_t)i * 2 * kD,
                    nullptr, nullptr, o1, kBS, 2 * kD, 4 * kD, 0, ACT_GELU, stream);
        launch_gemm(o1, W_o2 + (size_t)i * 2 * kD * kD, b_o2 + i * kD,
                    x, h, x, kBS, kD, 2 * kD, 0, ACT_NONE, stream);
    }

    // final LN into h, then tied head: logits = h @ emb^T  (emb is [V, D])
    layernorm_kernel<<<kBS, 256, 0, stream>>>(x, lnf_g, lnf_b, h, kD);
    launch_gemm(h, emb, nullptr, nullptr, nullptr, out,
                kBS, kV, kD, /*transB=*/1, ACT_NONE, stream);
}